// GATLSTM_3736621547944
// MI455X (gfx1250) — compile-verified
//
#include <hip/hip_runtime.h>
#include <cstdint>

#define NN   100000
#define EE   1600000
#define CHN  200
#define DDIM 32
#define EDD  64
#define NEF_ 32
#define NCC  16
#define EPSF 1e-5f

typedef float v2f __attribute__((ext_vector_type(2)));
typedef float v8f __attribute__((ext_vector_type(8)));

__device__ __forceinline__ v8f wmma4(v2f a, v2f b, v8f c) {
  // D = A(16x4 f32) * B(4x16 f32) + C(16x16 f32)
  return __builtin_amdgcn_wmma_f32_16x16x4_f32(false, a, false, b, (short)0, c, false, false);
}

// sum across the 16 lanes of each half-wave (masks < 16 never cross halves)
__device__ __forceinline__ float hsum16(float v) {
  v += __shfl_xor(v, 1, 32);
  v += __shfl_xor(v, 2, 32);
  v += __shfl_xor(v, 4, 32);
  v += __shfl_xor(v, 8, 32);
  return v;
}

// ---------------- Kernel 0: repack weight [K,Ncols] into K-paired layout ----------------
// out[(k>>1)*2*Ncols + n*2 + (k&1)] = in[k*Ncols + n]  -> B fragments become one b64 load
__global__ void k_pack(const float* __restrict__ in, float* __restrict__ out,
                       int K, int Ncols) {
  int i = blockIdx.x * blockDim.x + threadIdx.x;
  if (i >= K * Ncols) return;
  int k = i / Ncols, n = i - k * Ncols;
  out[(size_t)(k >> 1) * (2 * Ncols) + n * 2 + (k & 1)] = in[i];
}

// ---------------- Kernel 1: per-chunk node projector: h = relu(LN(x_c @ Wc + bc)) ----------------
__global__ __launch_bounds__(128) void k_node_proj(
    const float* __restrict__ x,
    const float* __restrict__ Wn0, const float* __restrict__ bn0,
    const float* __restrict__ Wn1, const float* __restrict__ bn1,
    const float* __restrict__ Wn2, const float* __restrict__ bn2,
    float* __restrict__ hbuf)
{
  __shared__ float sW[CHN * DDIM];   // paired layout: [(k>>1)][n][k&1], 25.6 KB
  __shared__ float sB[DDIM];
  const int c = blockIdx.y;
  const float* W = (c == 0) ? Wn0 : ((c == 1) ? Wn1 : Wn2);
  const float* B = (c == 0) ? bn0 : ((c == 1) ? bn1 : bn2);
  for (int i = threadIdx.x; i < CHN * DDIM; i += blockDim.x) {
    int k = i >> 5, n = i & 31;                 // DDIM == 32
    sW[((k >> 1) * 64) + n * 2 + (k & 1)] = W[i];
  }
  if (threadIdx.x < DDIM) sB[threadIdx.x] = B[threadIdx.x];
  __syncthreads();

  const int lane = threadIdx.x & 31;
  const int wv   = threadIdx.x >> 5;
  const int half = lane >> 4;
  const int l15  = lane & 15;
  const int koff = 2 * half;
  const int base = blockIdx.x * 64 + wv * 16;
  int arow = base + l15; if (arow > NN - 1) arow = NN - 1;   // clamp, keep EXEC full
  const float* aptr = x + (size_t)arow * (3 * CHN) + c * CHN;

  // branch-free near-scope prefetch of this lane's A row (200 floats = 800 B)
#pragma unroll
  for (int pf = 0; pf < CHN; pf += 32) __builtin_prefetch(aptr + pf, 0, 3);

  v8f acc0 = {}; v8f acc1 = {};
  for (int kb = 0; kb < CHN; kb += 4) {
    v2f a = *(const v2f*)(aptr + kb + koff);
    const int kp = (kb + koff) >> 1;
    v2f b0 = *(const v2f*)&sW[kp * 64 + l15 * 2];         // ds_load_b64
    v2f b1 = *(const v2f*)&sW[kp * 64 + (16 + l15) * 2];
    acc0 = wmma4(a, b0, acc0);
    acc1 = wmma4(a, b1, acc1);
  }
  const float bias0 = sB[l15], bias1 = sB[16 + l15];
#pragma unroll
  for (int v = 0; v < 8; ++v) {
    float t0 = acc0[v] + bias0;
    float t1 = acc1[v] + bias1;
    float s = hsum16(t0 + t1);
    float q = hsum16(t0 * t0 + t1 * t1);
    float mean = s * (1.f / 32.f);
    float var  = q * (1.f / 32.f) - mean * mean;
    float rs   = rsqrtf(var + EPSF);
    float o0 = fmaxf((t0 - mean) * rs, 0.f);
    float o1 = fmaxf((t1 - mean) * rs, 0.f);
    int row = base + v + 8 * half;     // D layout: M = vgpr + 8*half
    if (row < NN) {
      hbuf[(size_t)row * 96 + c * DDIM + l15]      = o0;
      hbuf[(size_t)row * 96 + c * DDIM + 16 + l15] = o1;
    }
  }
}

// ---------------- Kernel 2: xl = h @ Wl, xr = h @ Wr  ([N,96] @ [96,128], paired weights) ----------------
__global__ __launch_bounds__(256) void k_xlxr(
    const float* __restrict__ hbuf, const float* __restrict__ pWl,
    const float* __restrict__ pWr, float* __restrict__ xl, float* __restrict__ xr)
{
  const int lane = threadIdx.x & 31;
  const int wv   = threadIdx.x >> 5;   // output N-tile 0..7
  const int half = lane >> 4;
  const int l15  = lane & 15;
  const int koff = 2 * half;
  const int base = blockIdx.x * 16;    // N % 16 == 0
  const int ncol = wv * 16 + l15;
  const float* aptr = hbuf + (size_t)(base + l15) * 96;

#pragma unroll
  for (int pf = 0; pf < 96; pf += 32) __builtin_prefetch(aptr + pf, 0, 3);

  v8f aL = {}; v8f aR = {};
  for (int kb = 0; kb < 96; kb += 4) {
    v2f a = *(const v2f*)(aptr + kb + koff);
    const int kp = (kb + koff) >> 1;
    v2f bl = *(const v2f*)&pWl[(size_t)kp * 256 + ncol * 2];  // global_load_b64
    v2f br = *(const v2f*)&pWr[(size_t)kp * 256 + ncol * 2];
    aL = wmma4(a, bl, aL);
    aR = wmma4(a, br, aR);
  }
#pragma unroll
  for (int v = 0; v < 8; ++v) {
    const int row = base + v + 8 * half;
    xl[(size_t)row * 128 + ncol] = aL[v];
    xr[(size_t)row * 128 + ncol] = aR[v];
  }
}

// ---------------- Kernel 3: fused edge path ----------------
// e = relu(LN(edge_attr @ We + be)); ee = e @ Weg (kept in regs);
// m = leaky_relu(xl[src]+xr[dst]+ee); alpha_raw = <m, att>
__global__ __launch_bounds__(256) void k_edge(
    const float* __restrict__ ea, const float* __restrict__ pWe,
    const float* __restrict__ be, const float* __restrict__ pWeg,
    const float* __restrict__ att, const float* __restrict__ xl,
    const float* __restrict__ xr, const int* __restrict__ srcI,
    const int* __restrict__ dstI, float* __restrict__ araw)
{
  __shared__ float sE[8 * 16 * 68];    // per-wave 16x64 e-tile, stride 68 (34.8 KB)
  const int lane = threadIdx.x & 31;
  const int wv   = threadIdx.x >> 5;
  const int half = lane >> 4;
  const int l15  = lane & 15;
  const int koff = 2 * half;
  const int ebase = blockIdx.x * 128 + wv * 16;   // E % 128 == 0
  float* myE = sE + wv * (16 * 68);

  // --- GEMM1: edge projector [16,32] @ [32,64] ---
  const float* aptr = ea + (size_t)(ebase + l15) * NEF_;
  __builtin_prefetch(aptr, 0, 3);
  v8f eacc[4];
#pragma unroll
  for (int t = 0; t < 4; ++t) { v8f z = {}; eacc[t] = z; }
  for (int kb = 0; kb < NEF_; kb += 4) {
    v2f a = *(const v2f*)(aptr + kb + koff);
    const int kp = (kb + koff) >> 1;
#pragma unroll
    for (int t = 0; t < 4; ++t) {
      v2f b = *(const v2f*)&pWe[(size_t)kp * 128 + (t * 16 + l15) * 2];
      eacc[t] = wmma4(a, b, eacc[t]);
    }
  }
  float biasv[4];
#pragma unroll
  for (int t = 0; t < 4; ++t) biasv[t] = be[t * 16 + l15];

  // LN(64) + ReLU per edge row; store transposable tile to LDS
#pragma unroll
  for (int v = 0; v < 8; ++v) {
    float r0 = eacc[0][v] + biasv[0];
    float r1 = eacc[1][v] + biasv[1];
    float r2 = eacc[2][v] + biasv[2];
    float r3 = eacc[3][v] + biasv[3];
    float s = hsum16(r0 + r1 + r2 + r3);
    float q = hsum16(r0 * r0 + r1 * r1 + r2 * r2 + r3 * r3);
    float mean = s * (1.f / 64.f);
    float var  = q * (1.f / 64.f) - mean * mean;
    float rs   = rsqrtf(var + EPSF);
    const int m = v + 8 * half;
    myE[m * 68 +  0 + l15] = fmaxf((r0 - mean) * rs, 0.f);
    myE[m * 68 + 16 + l15] = fmaxf((r1 - mean) * rs, 0.f);
    myE[m * 68 + 32 + l15] = fmaxf((r2 - mean) * rs, 0.f);
    myE[m * 68 + 48 + l15] = fmaxf((r3 - mean) * rs, 0.f);
  }

  // --- GEMM2: ee = e-tile [16,64] @ Weg [64,128]; A from LDS (row-major re-read) ---
  v8f macc[8];
#pragma unroll
  for (int t = 0; t < 8; ++t) { v8f z = {}; macc[t] = z; }
  for (int kb = 0; kb < EDD; kb += 4) {
    v2f a = *(const v2f*)&myE[l15 * 68 + kb + koff];   // ds_load_b64
    const int kp = (kb + koff) >> 1;
#pragma unroll
    for (int t = 0; t < 8; ++t) {
      v2f b = *(const v2f*)&pWeg[(size_t)kp * 256 + (t * 16 + l15) * 2];
      macc[t] = wmma4(a, b, macc[t]);
    }
  }

  float attq[8];
#pragma unroll
  for (int t = 0; t < 8; ++t) attq[t] = att[(t >> 1) * 32 + (t & 1) * 16 + l15];

#pragma unroll
  for (int v = 0; v < 8; ++v) {
    const int eid = ebase + v + 8 * half;
    const int si = srcI[eid], di = dstI[eid];
    float p[8];
#pragma unroll
    for (int t = 0; t < 8; ++t) {
      const int n = t * 16 + l15;
      float val = macc[t][v] + xl[(size_t)si * 128 + n] + xr[(size_t)di * 128 + n];
      val = (val > 0.f) ? val : 0.2f * val;     // leaky_relu(0.2)
      p[t] = val * attq[t];
    }
#pragma unroll
    for (int hh = 0; hh < 4; ++hh) {
      float a = hsum16(p[2 * hh] + p[2 * hh + 1]);
      if (l15 == 0) araw[(size_t)eid * 4 + hh] = a;
    }
  }
}

// ---------------- segment softmax helpers ----------------
__global__ void k_amax_init(unsigned* __restrict__ amenc, int n) {
  int i = blockIdx.x * blockDim.x + threadIdx.x;
  if (i < n) amenc[i] = 0x007FFFFFu;  // order-encoding of -inf
}

__global__ void k_amax(const float* __restrict__ araw, const int* __restrict__ dstI,
                       unsigned* __restrict__ amenc, int n) {
  int i = blockIdx.x * blockDim.x + threadIdx.x;
  if (i >= n) return;
  int e = i >> 2, hh = i & 3;
  unsigned u = __float_as_uint(araw[i]);
  unsigned enc = u ^ ((unsigned)((int)u >> 31) | 0x80000000u);  // order-preserving
  atomicMax(&amenc[(size_t)dstI[e] * 4 + hh], enc);
}

__global__ void k_amax_fin(const unsigned* __restrict__ amenc, float* __restrict__ amaxf, int n) {
  int i = blockIdx.x * blockDim.x + threadIdx.x;
  if (i >= n) return;
  unsigned v = amenc[i];
  unsigned u = (v & 0x80000000u) ? (v ^ 0x80000000u) : ~v;
  float f = __uint_as_float(u);
  amaxf[i] = __builtin_isfinite(f) ? f : 0.0f;    // empty segments -> 0
}

__global__ void k_exp(const float* __restrict__ araw, const float* __restrict__ amaxf,
                      const int* __restrict__ dstI, float* __restrict__ exb,
                      float* __restrict__ denom, int n) {
  int i = blockIdx.x * blockDim.x + threadIdx.x;
  if (i >= n) return;
  int e = i >> 2, hh = i & 3;
  int d = dstI[e];
  float ex = expf(araw[i] - amaxf[(size_t)d * 4 + hh]);
  exb[i] = ex;
  atomicAdd(&denom[(size_t)d * 4 + hh], ex);
}

// ---------------- Kernel 6: message scatter: out[dst] += xl[src] * alpha ----------------
__global__ void k_scatter(const float* __restrict__ xl, const float* __restrict__ exb,
                          const float* __restrict__ denom, const int* __restrict__ srcI,
                          const int* __restrict__ dstI, float* __restrict__ outb, int nEdges) {
  int i = blockIdx.x * blockDim.x + threadIdx.x;
  int e = i >> 5, j = i & 31;            // 32 threads/edge, 4 channels each
  if (e >= nEdges) return;
  int si = srcI[e], di = dstI[e];
  int hh = j >> 3;
  float an = exb[(size_t)e * 4 + hh] / (denom[(size_t)di * 4 + hh] + 1e-16f);
  const float4 xv = *(const float4*)(xl + (size_t)si * 128 + j * 4);
  float* op = outb + (size_t)di * 128 + j * 4;
  atomicAdd(op + 0, xv.x * an);
  atomicAdd(op + 1, xv.y * an);
  atomicAdd(op + 2, xv.z * an);
  atomicAdd(op + 3, xv.w * an);
}

// ---------------- Kernel 7: global sum/sumsq of relu(out) ----------------
__global__ __launch_bounds__(256) void k_reduce(const float* __restrict__ outb,
                                                double* __restrict__ gsum, int n) {
  __shared__ double ss[256], sq[256];
  double ls = 0.0, lq = 0.0;
  int stride = gridDim.x * blockDim.x;
  for (int i = blockIdx.x * blockDim.x + threadIdx.x; i < n; i += stride) {
    float r = fmaxf(outb[i], 0.f);
    ls += (double)r;
    lq += (double)r * (double)r;
  }
  ss[threadIdx.x] = ls; sq[threadIdx.x] = lq;
  __syncthreads();
  for (int off = 128; off > 0; off >>= 1) {
    if (threadIdx.x < off) { ss[threadIdx.x] += ss[threadIdx.x + off]; sq[threadIdx.x] += sq[threadIdx.x + off]; }
    __syncthreads();
  }
  if (threadIdx.x == 0) {
    atomicAdd(&gsum[0], ss[0]);
    atomicAdd(&gsum[1], sq[0]);
  }
}

// ---------------- Kernel 8: global-LN(relu(out)) @ Wp + bp -> LN(16) -> ReLU ----------------
__global__ __launch_bounds__(256) void k_final(const float* __restrict__ outb,
    const double* __restrict__ gsum, const float* __restrict__ pWp,
    const float* __restrict__ bp, float* __restrict__ dout) {
  __shared__ float sWp[128 * NCC];   // paired layout, 8 KB
  for (int i = threadIdx.x; i < 128 * NCC; i += blockDim.x) sWp[i] = pWp[i];
  __syncthreads();

  const int lane = threadIdx.x & 31;
  const int wv   = threadIdx.x >> 5;
  const int base = blockIdx.x * 128 + wv * 16;
  if (base >= NN) return;                  // wave-uniform (N % 16 == 0), after barrier
  const int half = lane >> 4;
  const int l15  = lane & 15;
  const int koff = 2 * half;
  const double M = (double)NN * 128.0;
  const double mu_d = gsum[0] / M;
  const float mu = (float)mu_d;
  const float var = (float)(gsum[1] / M - mu_d * mu_d);
  const float rs = rsqrtf(var + EPSF);
  const float* aptr = outb + (size_t)(base + l15) * 128;

#pragma unroll
  for (int pf = 0; pf < 128; pf += 32) __builtin_prefetch(aptr + pf, 0, 3);

  v8f acc = {};
  for (int kb = 0; kb < 128; kb += 4) {
    const int kr = kb + koff;
    v2f a;
    a.x = (fmaxf(aptr[kr],     0.f) - mu) * rs;
    a.y = (fmaxf(aptr[kr + 1], 0.f) - mu) * rs;
    v2f b = *(const v2f*)&sWp[(kr >> 1) * 32 + l15 * 2];   // ds_load_b64
    acc = wmma4(a, b, acc);
  }
  const float bl = bp[l15];
#pragma unroll
  for (int v = 0; v < 8; ++v) {
    float t = acc[v] + bl;
    float s = hsum16(t);
    float q = hsum16(t * t);
    float mean = s * (1.f / 16.f);
    float var2 = q * (1.f / 16.f) - mean * mean;
    float rs2 = rsqrtf(var2 + EPSF);
    const int row = base + v + 8 * half;
    dout[(size_t)row * NCC + l15] = fmaxf((t - mean) * rs2, 0.f);
  }
}

extern "C" void kernel_launch(void* const* d_in, const int* in_sizes, int n_in,
                              void* d_out, int out_size, void* d_ws, size_t ws_size,
                              hipStream_t stream) {
  (void)in_sizes; (void)n_in; (void)out_size; (void)ws_size;
  const float* x   = (const float*)d_in[0];
  const float* ea  = (const float*)d_in[1];
  const int*   eix = (const int*)d_in[2];
  const float* Wn0 = (const float*)d_in[3];
  const float* bn0 = (const float*)d_in[4];
  const float* Wn1 = (const float*)d_in[5];
  const float* bn1 = (const float*)d_in[6];
  const float* Wn2 = (const float*)d_in[7];
  const float* bn2 = (const float*)d_in[8];
  const float* We  = (const float*)d_in[9];
  const float* be  = (const float*)d_in[10];
  const float* Wl  = (const float*)d_in[11];
  const float* Wr  = (const float*)d_in[12];
  const float* Weg = (const float*)d_in[13];
  const float* att = (const float*)d_in[14];
  const float* Wp  = (const float*)d_in[15];
  const float* bp  = (const float*)d_in[16];
  const int* srcI = eix;
  const int* dstI = eix + EE;

  char* w = (char*)d_ws;   // all chunk sizes are multiples of 256B
  float*    hbuf  = (float*)w;    w += (size_t)NN * 96  * 4;
  float*    xlb   = (float*)w;    w += (size_t)NN * 128 * 4;
  float*    xrb   = (float*)w;    w += (size_t)NN * 128 * 4;
  float*    araw  = (float*)w;    w += (size_t)EE * 4   * 4;
  float*    exb   = (float*)w;    w += (size_t)EE * 4   * 4;
  unsigned* amenc = (unsigned*)w; w += (size_t)NN * 4   * 4;
  float*    amaxf = (float*)w;    w += (size_t)NN * 4   * 4;
  float*    denom = (float*)w;    w += (size_t)NN * 4   * 4;
  float*    outb  = (float*)w;    w += (size_t)NN * 128 * 4;
  double*   gsum  = (double*)w;   w += 256;
  float*    pWl   = (float*)w;    w += (size_t)96  * 128 * 4;
  float*    pWr   = (float*)w;    w += (size_t)96  * 128 * 4;
  float*    pWe   = (float*)w;    w += (size_t)32  * 64  * 4;
  float*    pWeg  = (float*)w;    w += (size_t)64  * 128 * 4;
  float*    pWp   = (float*)w;    w += (size_t)128 * NCC * 4;

  hipMemsetAsync(denom, 0, (size_t)NN * 4 * 4, stream);
  hipMemsetAsync(outb,  0, (size_t)NN * 128 * 4, stream);
  hipMemsetAsync(gsum,  0, 16, stream);

  // one-shot weight repacks (K-paired layout for single-b64 B fragments)
  k_pack<<<(96 * 128 + 255) / 256, 256, 0, stream>>>(Wl,  pWl,  96, 128);
  k_pack<<<(96 * 128 + 255) / 256, 256, 0, stream>>>(Wr,  pWr,  96, 128);
  k_pack<<<(32 * 64  + 255) / 256, 256, 0, stream>>>(We,  pWe,  32, 64);
  k_pack<<<(64 * 128 + 255) / 256, 256, 0, stream>>>(Weg, pWeg, 64, 128);
  k_pack<<<(128 * NCC + 255) / 256, 256, 0, stream>>>(Wp, pWp, 128, NCC);

  k_node_proj<<<dim3((NN + 63) / 64, 3), 128, 0, stream>>>(x, Wn0, bn0, Wn1, bn1, Wn2, bn2, hbuf);
  k_xlxr<<<NN / 16, 256, 0, stream>>>(hbuf, pWl, pWr, xlb, xrb);
  k_edge<<<EE / 128, 256, 0, stream>>>(ea, pWe, be, pWeg, att, xlb, xrb, srcI, dstI, araw);
  k_amax_init<<<(NN * 4 + 255) / 256, 256, 0, stream>>>(amenc, NN * 4);
  k_amax<<<(EE * 4 + 255) / 256, 256, 0, stream>>>(araw, dstI, amenc, EE * 4);
  k_amax_fin<<<(NN * 4 + 255) / 256, 256, 0, stream>>>(amenc, amaxf, NN * 4);
  k_exp<<<(EE * 4 + 255) / 256, 256, 0, stream>>>(araw, amaxf, dstI, exb, denom, EE * 4);
  k_scatter<<<((size_t)EE * 32 + 255) / 256, 256, 0, stream>>>(xlb, exb, denom, srcI, dstI, outb, EE);
  k_reduce<<<2048, 256, 0, stream>>>(outb, gsum, NN * 128);
  k_final<<<(NN + 127) / 128, 256, 0, stream>>>(outb, gsum, pWp, bp, (float*)d_out);
}